// RateLossWeight_57836029608554
// MI455X (gfx1250) — compile-verified
//
#include <hip/hip_runtime.h>

#define FEAT 16
#define NOISEV 0.01f

typedef __attribute__((ext_vector_type(2))) float v2f;
typedef __attribute__((ext_vector_type(8))) float v8f;

// ---------------------------------------------------------------------------
// Kernel 1: zero the interference accumulator [N*FEAT] and the scalar slot.
// ---------------------------------------------------------------------------
__global__ void rl_zero_kernel(float* __restrict__ ws, int n_elems) {
    int i = blockIdx.x * blockDim.x + threadIdx.x;
    if (i < n_elems) ws[i] = 0.0f;
}

// ---------------------------------------------------------------------------
// Kernel 2: edge scatter.  16 lanes cooperate on one edge: lane f handles
// feature f.  Gather of allocs[src] is a coalesced 64B row; one f32 atomic
// per lane into the L2-resident interference buffer.  Edge index streams are
// prefetched ahead with global_prefetch_b8 (gfx1250 prefetch path).
// ---------------------------------------------------------------------------
__global__ void rl_edge_kernel(const float* __restrict__ allocs,
                               const float* __restrict__ csi_edge,
                               const int*   __restrict__ src,
                               const int*   __restrict__ dst,
                               float*       __restrict__ interf,
                               int n_edges) {
    int tid = blockIdx.x * blockDim.x + threadIdx.x;
    int e   = tid >> 4;   // 16 lanes per edge
    int f   = tid & 15;
    if (e >= n_edges) return;

    // stream-ahead prefetch of the edge arrays (one lane per edge does it)
    if (f == 0) {
        int pe = e + 4096;
        if (pe < n_edges) {
            __builtin_prefetch(&src[pe], 0, 1);
            __builtin_prefetch(&dst[pe], 0, 1);
            __builtin_prefetch(&csi_edge[pe], 0, 1);
        }
    }

    int   s = src[e];
    int   d = dst[e];
    float c = csi_edge[e];
    float w = (s != d) ? (c * c) : 0.0f;

    float val = allocs[s * FEAT + f] * w;
    atomicAdd(&interf[d * FEAT + f], val);
}

// ---------------------------------------------------------------------------
// Kernel 3: node rate + weighted reduction.
// Each wave32 handles 16 nodes x 16 features.  Per-lane VALU/TRANS work
// builds a *weight-pre-multiplied* f32 A-tile (16x4 per WMMA); four chained
// V_WMMA_F32_16X16X4_F32 with an all-ones B matrix accumulate weighted
// 16-wide feature sums in the D accumulator.
//   log2(1 + c2*al/it) is computed division-free as
//   log2(it + c2*al) - log2(it)      (both args > 0 since it >= NOISE)
// A layout (ISA 7.12.2): lane (m,half): VGPR0=A[m,2*half], VGPR1=A[m,2*half+1].
// D layout: VGPR r -> M=r (lanes 0-15), M=8+r (lanes 16-31).
// Tail nodes are handled branchlessly: index clamped, weight zeroed, so EXEC
// stays all-ones through the WMMAs and the post-WMMA code is load-free.
// ---------------------------------------------------------------------------
__global__ void rl_node_kernel(const float* __restrict__ allocs,
                               const float* __restrict__ weights,
                               const float* __restrict__ csi_node,
                               const float* __restrict__ interf,
                               float*       __restrict__ accum,
                               int n_nodes) {
    const int lane = threadIdx.x & 31;
    const int wave = threadIdx.x >> 5;   // 8 waves per 256-thread block
    const int m    = lane & 15;
    const int half = lane >> 4;

    const int wave_base = blockIdx.x * 128 + wave * 16;  // 16 nodes per wave
    int node  = wave_base + m;
    int nclmp = node < n_nodes ? node : (n_nodes - 1);   // keep EXEC all-ones

    float csi = csi_node[nclmp];
    float c2  = csi * csi;
    float wt  = weights[nclmp];
    wt = (node < n_nodes) ? wt : 0.0f;   // branchless tail mask

    const float2* arow = (const float2*)(allocs + (size_t)nclmp * FEAT);
    const float2* irow = (const float2*)(interf + (size_t)nclmp * FEAT);

    v8f acc = {};                       // C/D accumulator (16x16 f32)
    #pragma unroll
    for (int chunk = 0; chunk < 4; ++chunk) {
        float2 al2 = arow[chunk * 2 + half];   // feats 4*chunk+2*half, +1
        float2 it2 = irow[chunk * 2 + half];
        float den0 = it2.x + NOISEV;
        float den1 = it2.y + NOISEV;
        float num0 = __fmaf_rn(c2, al2.x, den0);
        float num1 = __fmaf_rn(c2, al2.y, den1);
        v2f a;
        a[0] = wt * (__log2f(num0) - __log2f(den0));
        a[1] = wt * (__log2f(num1) - __log2f(den1));
        v2f b = {1.0f, 1.0f};           // all-ones B: D[m,n] += sum_k A[m,k]
        acc = __builtin_amdgcn_wmma_f32_16x16x4_f32(
                  /*neg_a=*/false, a, /*neg_b=*/false, b,
                  /*c_mod=*/(short)0, acc, /*reuse_a=*/false, /*reuse_b=*/false);
    }

    // Lanes with m==0 (lanes 0 and 16) hold D column N=0:
    //   lane 0:  acc[r] = weighted feature-sum of node wave_base + r
    //   lane 16: acc[r] = weighted feature-sum of node wave_base + 8 + r
    float part = 0.0f;
    if (m == 0) {
        part = ((acc[0] + acc[1]) + (acc[2] + acc[3])) +
               ((acc[4] + acc[5]) + (acc[6] + acc[7]));
    }
    // combine lane 0 + lane 16, then block-reduce
    part += __shfl(part, 16);

    __shared__ float sdata[8];
    if (lane == 0) sdata[wave] = part;
    __syncthreads();
    if (threadIdx.x == 0) {
        float s = ((sdata[0] + sdata[1]) + (sdata[2] + sdata[3])) +
                  ((sdata[4] + sdata[5]) + (sdata[6] + sdata[7]));
        atomicAdd(accum, s);
    }
}

// ---------------------------------------------------------------------------
// Kernel 4: finalize  out = -sum / (N*FEAT)
// ---------------------------------------------------------------------------
__global__ void rl_finalize_kernel(const float* __restrict__ accum,
                                   float* __restrict__ out, float inv_total) {
    if (threadIdx.x == 0 && blockIdx.x == 0) out[0] = -accum[0] * inv_total;
}

// ---------------------------------------------------------------------------
extern "C" void kernel_launch(void* const* d_in, const int* in_sizes, int n_in,
                              void* d_out, int out_size, void* d_ws, size_t ws_size,
                              hipStream_t stream) {
    const float* allocs   = (const float*)d_in[0];
    const float* weights  = (const float*)d_in[1];
    const float* csi_node = (const float*)d_in[2];
    const float* csi_edge = (const float*)d_in[3];
    const int*   src      = (const int*)d_in[4];
    const int*   dst      = (const int*)d_in[5];

    const int n_nodes = in_sizes[1];          // weights: [N]
    const int n_edges = in_sizes[3];          // csi_edge: [E]

    float* interf = (float*)d_ws;             // [N*FEAT] accumulator
    float* accum  = interf + (size_t)n_nodes * FEAT;  // scalar slot
    float* out    = (float*)d_out;

    // 1) zero interference + scalar accumulator
    int n_zero = n_nodes * FEAT + 1;
    rl_zero_kernel<<<(n_zero + 255) / 256, 256, 0, stream>>>(interf, n_zero);

    // 2) edge scatter: 16 lanes per edge
    long long n_threads = (long long)n_edges * FEAT;
    int e_blocks = (int)((n_threads + 255) / 256);
    rl_edge_kernel<<<e_blocks, 256, 0, stream>>>(allocs, csi_edge, src, dst,
                                                 interf, n_edges);

    // 3) node rate + WMMA weighted feature reduction: 128 nodes / block
    int n_blocks = (n_nodes + 127) / 128;
    rl_node_kernel<<<n_blocks, 256, 0, stream>>>(allocs, weights, csi_node,
                                                 interf, accum, n_nodes);

    // 4) finalize
    float inv_total = 1.0f / ((float)n_nodes * (float)FEAT);
    rl_finalize_kernel<<<1, 32, 0, stream>>>(accum, out, inv_total);
}